// mLSTM_43370579755363
// MI455X (gfx1250) — compile-verified
//
#include <hip/hip_runtime.h>
#include <hip/hip_bf16.h>
#include <math.h>

// ---------------------------------------------------------------------------
// mLSTM (T=8192, IN=512, H=M=128) — chunkwise-parallel scan on CDNA5 WMMA.
// Phase 1: q/k/v/o projections via V_WMMA_F32_16X16X4_F32 (fp32 precision).
// Phase 2: scalar gates i=exp(.), logf=logsigmoid(.).
// Phase 3: single persistent workgroup; chunk L=64; per chunk four WMMA GEMMs:
//   S = Q Kᵀ, H_intra = (W∘S) V, H_inter = Q Cᵀ, C += (u∘V)ᵀ K; C in LDS.
// ---------------------------------------------------------------------------

typedef __attribute__((ext_vector_type(2))) float v2f;
typedef __attribute__((ext_vector_type(8))) float v8f;

#define T_LEN 8192
#define IN_DIM 512
#define MD 128           // hidden == mem dim
#define LCH 64           // chunk length
#define NCH (T_LEN / LCH)

__device__ __forceinline__ v8f wmma4(v2f a, v2f b, v8f c) {
  // D = A(16x4, f32) * B(4x16, f32) + C(16x16, f32)
  return __builtin_amdgcn_wmma_f32_16x16x4_f32(
      /*neg_a=*/false, a, /*neg_b=*/false, b,
      /*c_mod=*/(short)0, c, /*reuse_a=*/false, /*reuse_b=*/false);
}

__device__ __forceinline__ float sigmoidf_(float z) {
  return 1.0f / (1.0f + __expf(-z));
}

// ---------------------------------------------------------------------------
// Kernel 1: projections. grid = (T/16, 8), block = 128 (4 waves).
// colTile = blockIdx.y*4 + wave in [0,32): matrix = colTile>>3, 16-col tile.
// ---------------------------------------------------------------------------
__global__ __launch_bounds__(128) void proj_kernel(
    const float* __restrict__ x,
    const float* __restrict__ Wq, const float* __restrict__ bq,
    const float* __restrict__ Wk, const float* __restrict__ bk,
    const float* __restrict__ Wv, const float* __restrict__ bv,
    const float* __restrict__ Wo, const float* __restrict__ bo,
    float* __restrict__ qw, float* __restrict__ kw,
    float* __restrict__ vw, float* __restrict__ ow) {
  const int wave = threadIdx.x >> 5;
  const int lane = threadIdx.x & 31;
  const int half = lane >> 4;
  const int lm = lane & 15;

  const int rowBase = blockIdx.x * 16;
  const int colTile = blockIdx.y * 4 + wave;   // 0..31
  const int mat = colTile >> 3;                // 0..3
  const int col = (colTile & 7) * 16;

  const float* W; const float* bias; float* dst;
  switch (mat) {
    case 0:  W = Wq; bias = bq; dst = qw; break;
    case 1:  W = Wk; bias = bk; dst = kw; break;
    case 2:  W = Wv; bias = bv; dst = vw; break;
    default: W = Wo; bias = bo; dst = ow; break;
  }

  v8f acc = {};
  const float* arow = x + (size_t)(rowBase + lm) * IN_DIM + 2 * half;
  const float* brow = W + (size_t)(col + lm) * IN_DIM + 2 * half;
  for (int kk = 0; kk < IN_DIM; kk += 4) {
    v2f a, b;
    a.x = arow[kk];     a.y = arow[kk + 1];
    b.x = brow[kk];     b.y = brow[kk + 1];
    acc = wmma4(a, b, acc);
  }

  const float bv_ = bias[col + lm];
#pragma unroll
  for (int e = 0; e < 8; ++e) {
    const int r = rowBase + e + 8 * half;
    float val = acc[e] + bv_;
    if (mat == 1) val *= 0.08838834764831845f;        // 1/sqrt(128)
    if (mat == 3) val = sigmoidf_(val);               // output gate
    dst[(size_t)r * MD + col + lm] = val;
  }
}

// ---------------------------------------------------------------------------
// Kernel 2: scalar gates. ig[t]=exp(x.wi+bi), lf[t]=log sigmoid(x.wf+bf).
// ---------------------------------------------------------------------------
__global__ __launch_bounds__(256) void gates_kernel(
    const float* __restrict__ x,
    const float* __restrict__ wi, const float* __restrict__ bi,
    const float* __restrict__ wf, const float* __restrict__ bf,
    float* __restrict__ ig, float* __restrict__ lf) {
  const int t = blockIdx.x * blockDim.x + threadIdx.x;
  if (t >= T_LEN) return;
  const float* xr = x + (size_t)t * IN_DIM;
  float si = 0.f, sf = 0.f;
  for (int f = 0; f < IN_DIM; ++f) {
    const float xv = xr[f];
    si = fmaf(xv, wi[f], si);
    sf = fmaf(xv, wf[f], sf);
  }
  ig[t] = __expf(si + bi[0]);
  const float z = sf + bf[0];
  // stable log sigmoid
  lf[t] = (z >= 0.f) ? -log1pf(__expf(-z)) : (z - log1pf(__expf(z)));
}

// ---------------------------------------------------------------------------
// Kernel 3: chunkwise recurrent scan. Single block, 256 threads (8 waves).
// ---------------------------------------------------------------------------
__global__ __launch_bounds__(256, 1) void scan_kernel(
    const float* __restrict__ qw, const float* __restrict__ kw,
    const float* __restrict__ vw, const float* __restrict__ ow,
    const float* __restrict__ ig, const float* __restrict__ lf,
    const float* __restrict__ C0, const float* __restrict__ n0,
    const float* __restrict__ fc_w, const float* __restrict__ fc_b,
    float* __restrict__ out, float* __restrict__ C_out,
    float* __restrict__ n_out) {
  __shared__ float Cs[MD][MD + 1];       // C state, padded (bank-conflict free)
  __shared__ float Ps[LCH][LCH + 1];     // P = W ∘ S
  __shared__ float Hs[LCH][MD + 1];      // gated hidden rows
  __shared__ float la_s[LCH], ig_s[LCH], u_s[LCH], rs_s[LCH], qn_s[LCH];
  __shared__ float ns[MD], fcw_s[MD];

  const int tid = threadIdx.x;
  const int wave = tid >> 5;
  const int lane = tid & 31;
  const int half = lane >> 4;
  const int lm = lane & 15;

  for (int i = tid; i < MD * MD; i += 256) Cs[i >> 7][i & 127] = C0[i];
  if (tid < MD) { ns[tid] = n0[tid]; fcw_s[tid] = fc_w[tid]; }
  __syncthreads();

  for (int c = 0; c < NCH; ++c) {
    const int t0 = c * LCH;

    // Prefetch next chunk's Q/K/V (global_prefetch_b8), one 128B line/thread.
    if (c + 1 < NCH) {
      const float* nq = qw + (size_t)(t0 + LCH) * MD + tid * 32;
      const float* nk = kw + (size_t)(t0 + LCH) * MD + tid * 32;
      const float* nv = vw + (size_t)(t0 + LCH) * MD + tid * 32;
      __builtin_prefetch(nq, 0, 3);
      __builtin_prefetch(nk, 0, 3);
      __builtin_prefetch(nv, 0, 3);
    }

    if (tid == 0) {  // within-chunk inclusive cumsum of log f
      float a = 0.f;
      for (int j = 0; j < LCH; ++j) {
        a += lf[t0 + j];
        la_s[j] = a;
        ig_s[j] = ig[t0 + j];
      }
    }
    __syncthreads();
    if (tid < LCH) {
      u_s[tid] = __expf(la_s[LCH - 1] - la_s[tid]) * ig_s[tid];
      float s = 0.f;  // q_t · n_prev
      const float* qrow = qw + (size_t)(t0 + tid) * MD;
      for (int f = 0; f < MD; ++f) s = fmaf(qrow[f], ns[f], s);
      qn_s[tid] = s;
    }
    __syncthreads();

    // ---- Phase S: S = Qc Kcᵀ ; P = causal(W ∘ S) -> Ps ----
    for (int tile = wave; tile < 16; tile += 8) {
      const int tr = (tile >> 2) * 16, tc = (tile & 3) * 16;
      v8f acc = {};
      const float* ap = qw + (size_t)(t0 + tr + lm) * MD + 2 * half;
      const float* bp = kw + (size_t)(t0 + tc + lm) * MD + 2 * half;
      for (int kk = 0; kk < MD; kk += 4) {
        v2f a, b;
        a.x = ap[kk]; a.y = ap[kk + 1];
        b.x = bp[kk]; b.y = bp[kk + 1];
        acc = wmma4(a, b, acc);
      }
#pragma unroll
      for (int e = 0; e < 8; ++e) {
        const int r = tr + e + 8 * half, cc = tc + lm;
        float p = 0.f;
        if (r >= cc) p = __expf(la_s[r] - la_s[cc]) * ig_s[cc] * acc[e];
        Ps[r][cc] = p;
      }
    }
    __syncthreads();
    if (tid < LCH) {  // row sums of P (normalizer contribution)
      float s = 0.f;
      for (int j = 0; j <= tid; ++j) s += Ps[tid][j];
      rs_s[tid] = s;
    }
    __syncthreads();

    // ---- Phase H: H_intra = P @ V (K=64); H_inter = Q @ Cᵀ (K=128) ----
    for (int tile = wave; tile < 32; tile += 8) {
      const int tr = (tile >> 3) * 16, tc = (tile & 7) * 16;
      v8f aci = {};
      for (int kk = 0; kk < LCH; kk += 4) {
        v2f a, b;
        const int j0 = kk + 2 * half;
        a.x = Ps[tr + lm][j0]; a.y = Ps[tr + lm][j0 + 1];
        const float* bp = vw + (size_t)(t0 + j0) * MD + tc + lm;
        b.x = bp[0]; b.y = bp[MD];
        aci = wmma4(a, b, aci);
      }
      v8f ace = {};
      const float* ap = qw + (size_t)(t0 + tr + lm) * MD + 2 * half;
      for (int kk = 0; kk < MD; kk += 4) {
        v2f a, b;
        a.x = ap[kk]; a.y = ap[kk + 1];
        b.x = Cs[tc + lm][kk + 2 * half];
        b.y = Cs[tc + lm][kk + 2 * half + 1];
        ace = wmma4(a, b, ace);
      }
#pragma unroll
      for (int e = 0; e < 8; ++e) {
        const int r = tr + e + 8 * half, cc = tc + lm;
        const float Ft = __expf(la_s[r]);
        const float num = fmaf(Ft, ace[e], aci[e]);
        const float den = fmaxf(fabsf(fmaf(Ft, qn_s[r], rs_s[r])), 1.0f);
        Hs[r][cc] = (num / den) * ow[(size_t)(t0 + r) * MD + cc];
      }
    }
    __syncthreads();

    // out[t] = Hs[t] · fc_w + fc_b
    if (tid < LCH) {
      float s = 0.f;
      for (int f = 0; f < MD; ++f) s = fmaf(Hs[tid][f], fcw_s[f], s);
      out[t0 + tid] = s + fc_b[0];
    }

    // ---- State update: C = F_end*C + (u∘V)ᵀ K ; n = F_end*n + Kᵀ u ----
    const float Fend = __expf(la_s[LCH - 1]);
    for (int tile = wave; tile < 64; tile += 8) {
      const int tr = (tile >> 3) * 16, tc = (tile & 7) * 16;
      v8f acc;
#pragma unroll
      for (int e = 0; e < 8; ++e) acc[e] = Fend * Cs[tr + e + 8 * half][tc + lm];
      for (int kk = 0; kk < LCH; kk += 4) {
        v2f a, b;
        const int j0 = kk + 2 * half;
        a.x = u_s[j0]     * vw[(size_t)(t0 + j0) * MD + tr + lm];
        a.y = u_s[j0 + 1] * vw[(size_t)(t0 + j0 + 1) * MD + tr + lm];
        const float* bp = kw + (size_t)(t0 + j0) * MD + tc + lm;
        b.x = bp[0]; b.y = bp[MD];
        acc = wmma4(a, b, acc);
      }
#pragma unroll
      for (int e = 0; e < 8; ++e) Cs[tr + e + 8 * half][tc + lm] = acc[e];
    }
    if (tid < MD) {
      float s = Fend * ns[tid];
      for (int j = 0; j < LCH; ++j)
        s = fmaf(u_s[j], kw[(size_t)(t0 + j) * MD + tid], s);
      ns[tid] = s;
    }
    __syncthreads();
  }

  for (int i = tid; i < MD * MD; i += 256) C_out[i] = Cs[i >> 7][i & 127];
  if (tid < MD) n_out[tid] = ns[tid];
}

// ---------------------------------------------------------------------------
extern "C" void kernel_launch(void* const* d_in, const int* in_sizes, int n_in,
                              void* d_out, int out_size, void* d_ws, size_t ws_size,
                              hipStream_t stream) {
  const float* x    = (const float*)d_in[0];
  const float* C0   = (const float*)d_in[1];
  const float* n0   = (const float*)d_in[2];
  const float* Wq   = (const float*)d_in[3];
  const float* bq   = (const float*)d_in[4];
  const float* Wk   = (const float*)d_in[5];
  const float* bk   = (const float*)d_in[6];
  const float* Wv   = (const float*)d_in[7];
  const float* bv   = (const float*)d_in[8];
  const float* wi   = (const float*)d_in[9];
  const float* bi   = (const float*)d_in[10];
  const float* wf   = (const float*)d_in[11];
  const float* bf   = (const float*)d_in[12];
  const float* Wo   = (const float*)d_in[13];
  const float* bo   = (const float*)d_in[14];
  const float* fc_w = (const float*)d_in[15];
  const float* fc_b = (const float*)d_in[16];

  float* ws = (float*)d_ws;
  const size_t NP = (size_t)T_LEN * MD;   // 1,048,576 floats per projection
  float* qw = ws;
  float* kw = ws + NP;
  float* vw = ws + 2 * NP;
  float* owp = ws + 3 * NP;
  float* igp = ws + 4 * NP;
  float* lfp = igp + T_LEN;

  float* outp  = (float*)d_out;            // [T]
  float* Coutp = outp + T_LEN;             // [128*128]
  float* noutp = Coutp + MD * MD;          // [128]

  proj_kernel<<<dim3(T_LEN / 16, 8), 128, 0, stream>>>(
      x, Wq, bq, Wk, bk, Wv, bv, Wo, bo, qw, kw, vw, owp);
  gates_kernel<<<dim3(T_LEN / 256), 256, 0, stream>>>(
      x, wi, bi, wf, bf, igp, lfp);
  scan_kernel<<<1, 256, 0, stream>>>(
      qw, kw, vw, owp, igp, lfp, C0, n0, fc_w, fc_b, outp, Coutp, noutp);
}